// Cif_89404039233761
// MI455X (gfx1250) — compile-verified
//
#include <hip/hip_runtime.h>
#include <math.h>

// ---------------------------------------------------------------------------
// CIF (Continuous Integrate-and-Fire) for MI455X / gfx1250.
//
// Fusion: Conv1d(C,C,3)+Linear(C,1) collapse to a 768-tap correlation:
//   logit[b,t] = sum_{dt,ci} w_eff[dt*256+ci] * hs[b, t-1+dt, ci] + bias_eff
// alpha = sigmoid(logit). Then a per-batch sequential CIF scan writes cs.
// Memory-bound (~100 MB moved, ~4-5 us floor at 23.3 TB/s). Alpha logits
// via V_WMMA_F32_16X16X4_F32 (f32 keeps fire timing faithful).
// ---------------------------------------------------------------------------

typedef float v2f __attribute__((ext_vector_type(2)));
typedef float v8f __attribute__((ext_vector_type(8)));

#define HAVE_WMMA_F32X4 __has_builtin(__builtin_amdgcn_wmma_f32_16x16x4_f32)

constexpr int   Bsz = 16;
constexpr int   Tsz = 2048;
constexpr int   Csz = 256;
constexpr float TH  = 1.0f;

// ---------------------------------------------------------------------------
// 1) Precompute w_eff[768] (= lin_w . conv_w folded), bias_eff, zero loss slot
// ---------------------------------------------------------------------------
__global__ void prep_kernel(const float* __restrict__ conv_w,  // [C][C][3]
                            const float* __restrict__ conv_b,  // [C]
                            const float* __restrict__ lin_w,   // [1][C]
                            const float* __restrict__ lin_b,   // [1]
                            float* __restrict__ w_eff,         // [3*256]
                            float* __restrict__ bias_eff,      // [1]
                            float* __restrict__ loss_slot) {
  int r = blockIdx.x * 256 + threadIdx.x;      // 0..767 ; r = dt*256 + ci
  if (r < 3 * Csz) {
    int ci = r & (Csz - 1);
    int dt = r >> 8;
    float s = 0.0f;
    for (int c = 0; c < Csz; ++c)
      s = fmaf(lin_w[c], conv_w[c * (Csz * 3) + ci * 3 + dt], s);
    w_eff[r] = s;
  }
  if (r == 0) {
    float s = lin_b[0];
    for (int c = 0; c < Csz; ++c) s = fmaf(lin_w[c], conv_b[c], s);
    *bias_eff = s;
    *loss_slot = 0.0f;
  }
}

// ---------------------------------------------------------------------------
// 2) Zero cs region (float4 grid-stride)
// ---------------------------------------------------------------------------
__global__ void zero_kernel(float4* __restrict__ p, int n4) {
  int i = blockIdx.x * blockDim.x + threadIdx.x;
  int stride = gridDim.x * blockDim.x;
  float4 z; z.x = z.y = z.z = z.w = 0.0f;
  for (; i < n4; i += stride) p[i] = z;
}

// ---------------------------------------------------------------------------
// 3) alpha = sigmoid(correlation) via WMMA f32 16x16x4.
//    One wave = one (b, 16-t) tile. A: 16 t-rows x K4 slice of the 768
//    reduction. B: w_eff in column N=0, zeros elsewhere. 192 WMMAs/tile.
//    Edge rows: the row pointer is clamped to a VALID row and the loaded
//    VALUE is zero-selected (v_cndmask) -> unconditional b64 loads, no
//    exec-toggle branches, loads pipeline across the unrolled iterations.
// ---------------------------------------------------------------------------
#if HAVE_WMMA_F32X4
__global__ void alpha_wmma_kernel(const float* __restrict__ hs,     // [B][T][C]
                                  const float* __restrict__ w_eff,  // [768]
                                  const float* __restrict__ bias_eff,
                                  float* __restrict__ alpha,        // [B][T]
                                  float* __restrict__ loss_slot) {
  const int lane = threadIdx.x & 31;
  const int wave = threadIdx.x >> 5;
  const int tile = blockIdx.x * 8 + wave;      // 0 .. B*T/16-1 = 2047
  const int b    = tile >> 7;                  // T/16 = 128 tiles per batch
  const int t0   = (tile & 127) << 4;
  const int M    = lane & 15;                  // A-matrix row (t offset)
  const int half = lane >> 4;                  // K sub-pair selector
  const int kb   = half * 2;                   // K offset within 4-slice

  const float* hsb = hs + (size_t)b * Tsz * Csz;
  const int    tb  = t0 + M - 1;               // window start row for this lane

  v8f acc = {0.f, 0.f, 0.f, 0.f, 0.f, 0.f, 0.f, 0.f};

  for (int dt = 0; dt < 3; ++dt) {
    const int  tt = tb + dt;
    const bool ok = (tt >= 0) && (tt < Tsz);
    // Clamp to a valid row: address always dereferenceable, value selected.
    const float* rp = hsb + (size_t)(ok ? tt : 0) * Csz + kb;
    const float* wp = w_eff + dt * Csz + kb;
#pragma unroll 8
    for (int j = 0; j < Csz / 4; ++j) {        // 64 K-slices of 4
      v2f av = *(const v2f*)(rp + j * 4);      // unconditional global_load_b64
      v2f wv = *(const v2f*)(wp + j * 4);      // uniform per half-wave
      v2f a, bv;
      a.x  = ok ? av.x : 0.0f;                 // v_cndmask, no branch
      a.y  = ok ? av.y : 0.0f;
      bv.x = (M == 0) ? wv.x : 0.0f;           // B: only column N==0 nonzero
      bv.y = (M == 0) ? wv.y : 0.0f;
      acc = __builtin_amdgcn_wmma_f32_16x16x4_f32(
          /*neg_a=*/false, a, /*neg_b=*/false, bv,
          /*c_mod=*/(short)0, acc, /*reuse_a=*/false, /*reuse_b=*/false);
    }
  }

  const float bias = *bias_eff;
  if (M == 0) {                                // lanes 0 (M=0..7), 16 (M=8..15)
    const int mb = half * 8;
    float* ap = alpha + (size_t)b * Tsz + t0 + mb;
    float s = 0.0f;
#pragma unroll
    for (int v = 0; v < 8; ++v) {
      float z  = acc[v] + bias;
      float al = 1.0f / (1.0f + __expf(-z));
      ap[v] = al;
      s += al;
    }
    atomicAdd(loss_slot, s);                   // loss_pen = sum of all alpha
  }
}
#else
// Fallback (if f32 16x16x4 builtin is unavailable): plain VALU dot product.
__global__ void alpha_valu_kernel(const float* __restrict__ hs,
                                  const float* __restrict__ w_eff,
                                  const float* __restrict__ bias_eff,
                                  float* __restrict__ alpha,
                                  float* __restrict__ loss_slot) {
  int g = blockIdx.x * blockDim.x + threadIdx.x;   // 0 .. B*T-1
  int b = g >> 11, t = g & (Tsz - 1);
  const float* hsb = hs + (size_t)b * Tsz * Csz;
  float s = 0.0f;
  for (int dt = 0; dt < 3; ++dt) {
    int tt = t - 1 + dt;
    if (tt < 0 || tt >= Tsz) continue;
    const float* rp = hsb + (size_t)tt * Csz;
    const float* wp = w_eff + dt * Csz;
    for (int ci = 0; ci < Csz; ++ci) s = fmaf(rp[ci], wp[ci], s);
  }
  float al = 1.0f / (1.0f + __expf(-(s + *bias_eff)));
  alpha[g] = al;
  atomicAdd(loss_slot, al);
}
#endif

// ---------------------------------------------------------------------------
// 4) CIF scan. Fire schedule depends only on alpha -> uniform per block, so
//    channels are split across blocks: grid (B, C/32), 1 wave per block.
//    alpha[b,:] staged in LDS; hs loads coalesced, 1-step register prefetch
//    plus an 8-row global_prefetch_b8 to deepen the memory pipeline across
//    the sequential dependence chain. Each cs slot gets at most one store
//    onto the pre-zeroed buffer.
// ---------------------------------------------------------------------------
__global__ void cif_scan_kernel(const float* __restrict__ hs,     // [B][T][C]
                                const float* __restrict__ alpha,  // [B][T]
                                float* __restrict__ cs) {         // [B][T][C]
  __shared__ float sA[Tsz];
  const int b = blockIdx.x;
  const int c = blockIdx.y * 32 + threadIdx.x;

  const float* ab = alpha + (size_t)b * Tsz;
  for (int i = threadIdx.x; i < Tsz; i += 32) sA[i] = ab[i];
  __syncthreads();

  const float* hp = hs + (size_t)b * Tsz * Csz + c;
  float*       op = cs + (size_t)b * Tsz * Csz + c;

  float aa    = sA[0];
  float hacc  = aa * hp[0];
  int   idx   = 0;
  float hnext = hp[Csz];                        // prefetch t=1

  for (int t = 1; t < Tsz; ++t) {
    // Deep prefetch (speculative, clamped): gfx1250 global_prefetch_b8.
    int tp = (t + 8 < Tsz) ? (t + 8) : (Tsz - 1);
    __builtin_prefetch(hp + (size_t)tp * Csz, 0, 1);

    float hcur = hnext;
    if (t + 1 < Tsz) hnext = hp[(size_t)(t + 1) * Csz];
    float au     = sA[t];
    float aa_new = aa + au;
    if (aa_new >= TH) {                         // uniform branch (alpha-only)
      float a1 = TH - aa;
      op[(size_t)idx * Csz] = fmaf(a1, hcur, hacc);
      float rem = au - a1;
      aa   = rem;
      hacc = rem * hcur;
      ++idx;
    } else {
      aa   = aa_new;
      hacc = fmaf(au, hcur, hacc);
    }
  }
  if (aa >= 0.5f * TH) op[(size_t)idx * Csz] = hacc;   // tail fire
}

// ---------------------------------------------------------------------------
extern "C" void kernel_launch(void* const* d_in, const int* in_sizes, int n_in,
                              void* d_out, int out_size, void* d_ws, size_t ws_size,
                              hipStream_t stream) {
  const float* hs     = (const float*)d_in[0];   // hs_pad [B,T,C]
  // d_in[1] = hs_mask (unused by the math)
  const float* conv_w = (const float*)d_in[2];   // [C,C,3]
  const float* conv_b = (const float*)d_in[3];   // [C]
  const float* lin_w  = (const float*)d_in[4];   // [1,C]
  const float* lin_b  = (const float*)d_in[5];   // [1]

  float* ws       = (float*)d_ws;
  float* w_eff    = ws;              // 768 floats
  float* bias_eff = ws + 768;        // 1 float
  float* alpha    = ws + 1024;       // B*T = 32768 floats

  float* cs   = (float*)d_out;                         // [B,T,C]
  float* loss = cs + (size_t)Bsz * Tsz * Csz;          // scalar loss_pen

  prep_kernel<<<3, 256, 0, stream>>>(conv_w, conv_b, lin_w, lin_b,
                                     w_eff, bias_eff, loss);

  zero_kernel<<<2048, 256, 0, stream>>>((float4*)cs, (Bsz * Tsz * Csz) / 4);

#if HAVE_WMMA_F32X4
  // B*T/16 = 2048 tiles, 8 waves (tiles) per 256-thread block -> 256 blocks
  alpha_wmma_kernel<<<256, 256, 0, stream>>>(hs, w_eff, bias_eff, alpha, loss);
#else
  alpha_valu_kernel<<<(Bsz * Tsz) / 256, 256, 0, stream>>>(hs, w_eff, bias_eff,
                                                           alpha, loss);
#endif

  cif_scan_kernel<<<dim3(Bsz, Csz / 32), 32, 0, stream>>>(hs, alpha, cs);
}